// Variance_Loss_64811056496910
// MI455X (gfx1250) — compile-verified
//
#include <hip/hip_runtime.h>

// ---------------------------------------------------------------------------
// Variance-of-patches loss for MI455X (gfx1250, wave32).
//   pred: [512, 240, 240] f32   gt: [240, 240] f32   out: scalar f32
// Memory-bound: 118MB @ 23.3TB/s => ~5us floor. Single pass over pred using
// per-patch (S1, S2); variance reconstructed as (S2 - 2*m*S1)/N + m^2.
// Cross-lane reductions use V_WMMA_F32_16X16X4_F32 with a ones A-matrix.
// ---------------------------------------------------------------------------

typedef float v2f __attribute__((ext_vector_type(2)));
typedef float v8f __attribute__((ext_vector_type(8)));

#define PS      20          // patch size
#define GS      12          // grid size (240/20)
#define NPATCH  144         // GS*GS
#define BATCH   512
#define SPLIT   8           // batch splits per patch
#define BPB     (BATCH / SPLIT)
#define THREADS 256
#define ROWF    240         // floats per image row
#define IMGF    (240 * 240) // floats per image

// Reduce one float per lane across a fully-active wave32 to a wave total.
// WMMA: D = ones(16x4) x B, so every D element in column n is
//       c[n] = B[:,n] summed = v[n] + v[n+16]   (b[1] rows are zero).
// d[0] lane l then holds c[l % 16]; 4 shfl_xor steps finish 16 -> 1.
__device__ __forceinline__ float wave_reduce_wmma(float v) {
    v2f a; a[0] = 1.0f; a[1] = 1.0f;   // A = ones (16x4)
    v2f b; b[0] = v;    b[1] = 0.0f;   // 32 lane partials live in B's VGPR0
    v8f c = {};
    c = __builtin_amdgcn_wmma_f32_16x16x4_f32(
            /*neg_a=*/false, a, /*neg_b=*/false, b,
            /*c_mod=*/(short)0, c, /*reuse_a=*/false, /*reuse_b=*/false);
    float s = c[0];                    // lane l: column sum c[l & 15]
    s += __shfl_xor(s, 1, 32);
    s += __shfl_xor(s, 2, 32);
    s += __shfl_xor(s, 4, 32);
    s += __shfl_xor(s, 8, 32);
    return s;                          // total of all 32 lane inputs
}

// Kernel 1: per-(patch, batch-split) sums of x and x^2 over the 20x20 strip.
// Grid: NPATCH * SPLIT blocks of 256 threads. Each block streams
// BPB*20*5 = 6400 float4 (102.4 KB). ws[blk] = {S1, S2}.
__global__ void vl_patch_sums(const float* __restrict__ pred,
                              float* __restrict__ ws) {
    const int blk   = blockIdx.x;
    const int patch = blk / SPLIT;
    const int split = blk % SPLIT;
    const int gy = patch / GS;
    const int gx = patch % GS;

    const float* base = pred + (size_t)(split * BPB) * IMGF
                             + (size_t)(gy * PS) * ROWF
                             + (size_t)(gx * PS);

    float s1 = 0.0f, s2 = 0.0f;
    const int total4 = BPB * PS * 5;   // float4 tiles in this block's strip
    for (int i = threadIdx.x; i < total4; i += THREADS) {
        const int f = i % 5;           // float4 within the 20-wide row seg
        const int r = i / 5;
        const int y = r % PS;
        const int b = r / PS;
        const float4 v = *reinterpret_cast<const float4*>(
            base + (size_t)b * IMGF + (size_t)y * ROWF + f * 4);
        s1 += v.x + v.y + v.z + v.w;
        s2 += v.x * v.x + v.y * v.y + v.z * v.z + v.w * v.w;
    }

    __shared__ float l1[THREADS];
    __shared__ float l2[THREADS];
    l1[threadIdx.x] = s1;
    l2[threadIdx.x] = s2;
    __syncthreads();

    if (threadIdx.x < 32) {            // wave 0, EXEC all ones
        float p1 = 0.0f, p2 = 0.0f;
        #pragma unroll
        for (int k = 0; k < THREADS / 32; ++k) {
            p1 += l1[threadIdx.x + k * 32];
            p2 += l2[threadIdx.x + k * 32];
        }
        const float t1 = wave_reduce_wmma(p1);
        const float t2 = wave_reduce_wmma(p2);
        if (threadIdx.x == 0) {
            ws[blk * 2 + 0] = t1;
            ws[blk * 2 + 1] = t2;
        }
    }
}

// Kernel 2: single block. Fold splits, derive global mean m, do gt patch
// sums (tiny: 230 KB), assemble the scalar loss.
__global__ void vl_finalize(const float* __restrict__ ws,
                            const float* __restrict__ gt,
                            float* __restrict__ out) {
    __shared__ float red[THREADS];
    __shared__ float m_sh;
    const int t = threadIdx.x;

    float S1 = 0.0f, S2 = 0.0f;
    if (t < NPATCH) {
        #pragma unroll
        for (int s = 0; s < SPLIT; ++s) {
            S1 += ws[(t * SPLIT + s) * 2 + 0];
            S2 += ws[(t * SPLIT + s) * 2 + 1];
        }
    }

    // Global mean of pred from the patch S1's.
    red[t] = (t < NPATCH) ? S1 : 0.0f;
    __syncthreads();
    for (int s = THREADS / 2; s > 0; s >>= 1) {
        if (t < s) red[t] += red[t + s];
        __syncthreads();
    }
    if (t == 0) m_sh = red[0] / (512.0f * 240.0f * 240.0f);
    __syncthreads();
    const float m = m_sh;

    float term = 0.0f;
    if (t < NPATCH) {
        const int gy = t / GS;
        const int gx = t % GS;
        const float* gbase = gt + (size_t)(gy * PS) * ROWF + gx * PS;
        float G1 = 0.0f, G2 = 0.0f;
        for (int y = 0; y < PS; ++y) {
            const float4* row =
                reinterpret_cast<const float4*>(gbase + (size_t)y * ROWF);
            #pragma unroll
            for (int f = 0; f < 5; ++f) {
                const float4 v = row[f];
                G1 += v.x + v.y + v.z + v.w;
                G2 += v.x * v.x + v.y * v.y + v.z * v.z + v.w * v.w;
            }
        }
        const float Np = 512.0f * 400.0f;
        const float pred_var = (S2 - 2.0f * m * S1) / Np + m * m;
        const float gt_var   = (G2 - 2.0f * m * G1) / 400.0f + m * m;
        const float d = pred_var - gt_var;
        term = d * d;
    }

    red[t] = term;
    __syncthreads();
    for (int s = THREADS / 2; s > 0; s >>= 1) {
        if (t < s) red[t] += red[t + s];
        __syncthreads();
    }
    if (t == 0) out[0] = red[0] * 0.5f / 12.0f;
}

extern "C" void kernel_launch(void* const* d_in, const int* in_sizes, int n_in,
                              void* d_out, int out_size, void* d_ws, size_t ws_size,
                              hipStream_t stream) {
    (void)in_sizes; (void)n_in; (void)out_size; (void)ws_size;
    const float* pred = (const float*)d_in[0];
    const float* gt   = (const float*)d_in[1];
    float* out = (float*)d_out;
    float* ws  = (float*)d_ws;   // NPATCH*SPLIT*2 floats = 9216 B, fully rewritten

    vl_patch_sums<<<NPATCH * SPLIT, THREADS, 0, stream>>>(pred, ws);
    vl_finalize<<<1, THREADS, 0, stream>>>(ws, gt, out);
}